// ExplicitNReadChain_64123861729639
// MI455X (gfx1250) — compile-verified
//
#include <hip/hip_runtime.h>
#include <math.h>

#define B_ 64
#define S_ 4096
#define D_ 256
#define E_ 32000
#define H_ 5
#define EPS_ 1e-12f
#define CH_ 8
#define SCHUNK_ (S_ / CH_)   // 512
#define NPART_ (CH_ * 4)     // 32 partials per (b,d)

typedef float v2f __attribute__((ext_vector_type(2)));
typedef float v4f __attribute__((ext_vector_type(4)));
typedef float v8f __attribute__((ext_vector_type(8)));

// LDS swizzle for the state tile: keeps v2f/v4f contiguity (XOR with multiple of 4),
// spreads the 16 A-fragment rows across distinct banks.
__device__ __forceinline__ int sw_idx(int row, int kk) {
    return row * D_ + (kk ^ ((row & 15) << 2));
}

// ---------------- simple copy ----------------
__global__ void nrc_copy_kernel(const float* __restrict__ src, float* __restrict__ dst, int n) {
    int i = blockIdx.x * blockDim.x + threadIdx.x;
    if (i < n) dst[i] = src[i];
}

// ---------------- hop step 1: q = state@Wq[h]+bq, L2-normalize ----------------
// grid = B blocks, 256 threads (thread = output dim d)
__global__ void nrc_qproj_kernel(const float* __restrict__ state,
                                 const float* __restrict__ Wq,
                                 const float* __restrict__ bq,
                                 float* __restrict__ qn, int hop) {
    __shared__ float srow[D_];
    __shared__ float red[256];
    const int b = blockIdx.x;
    const int d = threadIdx.x;
    srow[d] = state[b * D_ + d];
    __syncthreads();
    const float* W = Wq + (size_t)hop * D_ * D_;
    float acc = bq[hop * D_ + d];
#pragma unroll 8
    for (int k = 0; k < D_; ++k) acc = fmaf(srow[k], W[(size_t)k * D_ + d], acc);
    red[d] = acc * acc;
    __syncthreads();
    for (int off = 128; off > 0; off >>= 1) {
        if (d < off) red[d] += red[d + off];
        __syncthreads();
    }
    const float inv = 1.0f / fmaxf(sqrtf(red[0]), EPS_);
    qn[b * D_ + d] = acc * inv;
}

// ---------------- hop step 2: cosine logits (keys streamed once) ----------------
// block = 256 threads = 8 waves; wave handles one s row. grid = (S/8, B)
__global__ void nrc_logits_kernel(const float* __restrict__ keys,
                                  const float* __restrict__ qn,
                                  float* __restrict__ logits) {
    const int b    = blockIdx.y;
    const int wave = threadIdx.x >> 5;
    const int lane = threadIdx.x & 31;
    const int s    = blockIdx.x * 8 + wave;
    const float* krow = keys + ((size_t)b * S_ + s) * D_;
    const float* qrow = qn + (size_t)b * D_;
    const int base = lane * 8;
    __builtin_prefetch(krow + (size_t)8 * D_ + base, 0, 0);
    v4f k0 = *(const v4f*)(krow + base);
    v4f k1 = *(const v4f*)(krow + base + 4);
    v4f q0 = *(const v4f*)(qrow + base);
    v4f q1 = *(const v4f*)(qrow + base + 4);
    float dot = k0.x * q0.x + k0.y * q0.y + k0.z * q0.z + k0.w * q0.w
              + k1.x * q1.x + k1.y * q1.y + k1.z * q1.z + k1.w * q1.w;
    float ss  = k0.x * k0.x + k0.y * k0.y + k0.z * k0.z + k0.w * k0.w
              + k1.x * k1.x + k1.y * k1.y + k1.z * k1.z + k1.w * k1.w;
#pragma unroll
    for (int off = 16; off > 0; off >>= 1) {
        dot += __shfl_xor(dot, off, 32);
        ss  += __shfl_xor(ss, off, 32);
    }
    if (lane == 0)
        logits[(size_t)b * S_ + s] = dot / fmaxf(sqrtf(ss), EPS_);
}

// ---------------- hop step 3: per-row max and 1/sum(exp) ----------------
// grid = B blocks, 256 threads
__global__ void nrc_rowstat_kernel(const float* __restrict__ logits,
                                   float* __restrict__ m, float* __restrict__ invZ) {
    __shared__ float red[256];
    const int b = blockIdx.x;
    const int t = threadIdx.x;
    const float* row = logits + (size_t)b * S_;
    float mx = -INFINITY;
    for (int s = t; s < S_; s += 256) mx = fmaxf(mx, row[s]);
    red[t] = mx;
    __syncthreads();
    for (int off = 128; off > 0; off >>= 1) {
        if (t < off) red[t] = fmaxf(red[t], red[t + off]);
        __syncthreads();
    }
    mx = red[0];
    __syncthreads();
    float z = 0.0f;
    for (int s = t; s < S_; s += 256) z += expf(row[s] - mx);
    red[t] = z;
    __syncthreads();
    for (int off = 128; off > 0; off >>= 1) {
        if (t < off) red[t] += red[t + off];
        __syncthreads();
    }
    if (t == 0) { m[b] = mx; invZ[b] = 1.0f / red[0]; }
}

// ---------------- hop step 4: attn out + partial read (values streamed once) ----
// grid = (CH, B); block = 256 threads = 4 s-phases x 64 d-quads; b128 value loads
__global__ void nrc_attnread_kernel(const float* __restrict__ values,
                                    const float* __restrict__ logits,
                                    const float* __restrict__ m,
                                    const float* __restrict__ invZ,
                                    float* __restrict__ attn_out,
                                    float* __restrict__ partial) {
    __shared__ float a[SCHUNK_];
    const int b = blockIdx.y;
    const int c = blockIdx.x;
    const int t = threadIdx.x;
    const int s0 = c * SCHUNK_;
    const float mb = m[b];
    const float iz = invZ[b];
    for (int i = t; i < SCHUNK_; i += 256) {
        const float av = expf(logits[(size_t)b * S_ + s0 + i] - mb) * iz;
        a[i] = av;
        attn_out[(size_t)b * S_ + s0 + i] = av;
    }
    __syncthreads();
    const int g = t >> 6;   // s phase 0..3
    const int u = t & 63;   // d quad 0..63
    const float* vbase = values + ((size_t)b * S_ + s0 + g) * D_ + u * 4;
    v4f acc = {0.0f, 0.0f, 0.0f, 0.0f};
#pragma unroll 4
    for (int j = 0; j < SCHUNK_ / 4; ++j) {
        __builtin_prefetch(vbase + (size_t)4 * (j + 8) * D_, 0, 0);
        v4f v = *(const v4f*)(vbase + (size_t)4 * j * D_);
        const float av = a[4 * j + g];
        acc.x = fmaf(av, v.x, acc.x);
        acc.y = fmaf(av, v.y, acc.y);
        acc.z = fmaf(av, v.z, acc.z);
        acc.w = fmaf(av, v.w, acc.w);
    }
    *(v4f*)(partial + (((size_t)(b * CH_ + c)) * 4 + g) * D_ + u * 4) = acc;
}

// ---------------- hop step 5: state += sum of partials ----------------
// grid = B blocks, 256 threads
__global__ void nrc_update_kernel(const float* __restrict__ partial,
                                  float* __restrict__ state) {
    const int b = blockIdx.x;
    const int t = threadIdx.x;
    float acc = state[(size_t)b * D_ + t];
#pragma unroll
    for (int p = 0; p < NPART_; ++p)
        acc += partial[((size_t)b * NPART_ + p) * D_ + t];
    state[(size_t)b * D_ + t] = acc;
}

// ---------------- hop step 6: logits_out = state @ Wd[h] + bd[h] (WMMA f32) ----
// One wave owns a 16-wide N tile and ALL 4 M tiles (B=64): the streamed Wd
// fragment feeds 4 WMMAs. state staged once per block in 64KB swizzled LDS.
// grid = E/16/8 = 250 blocks; block = 256 threads (8 waves -> 8 N tiles)
__global__ void __launch_bounds__(256)
nrc_wdgemm_kernel(const float* __restrict__ state,
                  const float* __restrict__ Wd,
                  const float* __restrict__ bd,
                  float* __restrict__ out, int hop) {
    __shared__ float slds[B_ * D_];   // exactly 64 KB, XOR-swizzled
    const int t = threadIdx.x;
    // cooperative staging: 16384 floats, v4f per thread per step
#pragma unroll
    for (int i = t; i < B_ * D_ / 4; i += 256) {
        const int r  = (i * 4) / D_;
        const int cc = (i * 4) % D_;
        *(v4f*)(slds + sw_idx(r, cc)) = *(const v4f*)(state + i * 4);
    }
    __syncthreads();

    const int wave = t >> 5;
    const int lane = t & 31;
    const int lclo = lane & 15;
    const int hi   = lane >> 4;                  // 0: lanes 0-15, 1: lanes 16-31
    const int n0   = (blockIdx.x * 8 + wave) * 16;
    const int col  = n0 + lclo;

    const float* W = Wd + (size_t)hop * D_ * E_;
    const float bias = bd[(size_t)hop * E_ + col];
    v8f c0 = {bias, bias, bias, bias, bias, bias, bias, bias};
    v8f c1 = c0, c2 = c0, c3 = c0;

    for (int k = 0; k < D_; k += 4) {
        const int kk = k + hi * 2;
        // B (4x16 f32): VGPR0 = rows K+0/K+2, VGPR1 = rows K+1/K+3
        v2f bm;
        bm.x = W[(size_t)(kk + 0) * E_ + col];
        bm.y = W[(size_t)(kk + 1) * E_ + col];
        // A fragments (16x4 f32) for the 4 M tiles, from swizzled LDS
        v2f a0 = *(const v2f*)(slds + sw_idx( 0 + lclo, kk));
        v2f a1 = *(const v2f*)(slds + sw_idx(16 + lclo, kk));
        v2f a2 = *(const v2f*)(slds + sw_idx(32 + lclo, kk));
        v2f a3 = *(const v2f*)(slds + sw_idx(48 + lclo, kk));
        c0 = __builtin_amdgcn_wmma_f32_16x16x4_f32(false, a0, false, bm, (short)0, c0, false, false);
        c1 = __builtin_amdgcn_wmma_f32_16x16x4_f32(false, a1, false, bm, (short)0, c1, false, false);
        c2 = __builtin_amdgcn_wmma_f32_16x16x4_f32(false, a2, false, bm, (short)0, c2, false, false);
        c3 = __builtin_amdgcn_wmma_f32_16x16x4_f32(false, a3, false, bm, (short)0, c3, false, false);
    }

    // D layout: VGPR v -> row mt*16 + v (lanes 0-15) / mt*16 + v + 8 (lanes 16-31)
    float* obase = out + (size_t)hop * B_ * E_;
#pragma unroll
    for (int v = 0; v < 8; ++v) {
        const int rr = hi * 8 + v;
        obase[(size_t)( 0 + rr) * E_ + col] = c0[v];
        obase[(size_t)(16 + rr) * E_ + col] = c1[v];
        obase[(size_t)(32 + rr) * E_ + col] = c2[v];
        obase[(size_t)(48 + rr) * E_ + col] = c3[v];
    }
}

extern "C" void kernel_launch(void* const* d_in, const int* in_sizes, int n_in,
                              void* d_out, int out_size, void* d_ws, size_t ws_size,
                              hipStream_t stream) {
    const float* query  = (const float*)d_in[0];
    const float* keys   = (const float*)d_in[1];
    const float* values = (const float*)d_in[2];
    const float* Wq     = (const float*)d_in[3];
    const float* bq     = (const float*)d_in[4];
    const float* Wd     = (const float*)d_in[5];
    const float* bd     = (const float*)d_in[6];

    float* out        = (float*)d_out;
    float* out_state  = out;                                    // [B,D]
    float* out_logits = out + B_ * D_;                          // [H,B,E]
    float* out_attn   = out + B_ * D_ + (size_t)H_ * B_ * E_;   // [H,B,S]

    float* ws      = (float*)d_ws;
    float* state   = ws;                            // B*D
    float* qn      = state + B_ * D_;               // B*D
    float* lg      = qn + B_ * D_;                  // B*S
    float* mrow    = lg + (size_t)B_ * S_;          // B
    float* izrow   = mrow + B_;                     // B
    float* partial = izrow + B_;                    // B*NPART*D (2 MB)

    nrc_copy_kernel<<<dim3(64), dim3(256), 0, stream>>>(query, state, B_ * D_);

    for (int h = 0; h < H_; ++h) {
        nrc_qproj_kernel<<<dim3(B_), dim3(256), 0, stream>>>(state, Wq, bq, qn, h);
        nrc_logits_kernel<<<dim3(S_ / 8, B_), dim3(256), 0, stream>>>(keys, qn, lg);
        nrc_rowstat_kernel<<<dim3(B_), dim3(256), 0, stream>>>(lg, mrow, izrow);
        nrc_attnread_kernel<<<dim3(CH_, B_), dim3(256), 0, stream>>>(
            values, lg, mrow, izrow, out_attn + (size_t)h * B_ * S_, partial);
        nrc_update_kernel<<<dim3(B_), dim3(256), 0, stream>>>(partial, state);
        nrc_wdgemm_kernel<<<dim3(E_ / 128), dim3(256), 0, stream>>>(
            state, Wd, bd, out_logits, h);
    }

    nrc_copy_kernel<<<dim3(64), dim3(256), 0, stream>>>(state, out_state, B_ * D_);
}